// NECNetwork_29867202576934
// MI455X (gfx1250) — compile-verified
//
#include <hip/hip_runtime.h>
#include <hip/hip_bf16.h>
#include <stdint.h>

typedef __attribute__((ext_vector_type(2))) float v2f;
typedef __attribute__((ext_vector_type(8))) float v8f;

#define NB   128
#define NIN  128
#define ND   32
#define NA   8
#define NC   100000
#define NP   50
#define NW   8
#define CAPK 50
#define SLAB 128   // candidates per LDS slab (16 KB of f32)

// d_out offsets (element units)
#define OFF_VALUES  0
#define OFF_ACTIONS 128
#define OFF_INDEXES 256
#define OFF_SCORES  51456
#define OFF_KEYS    102656

// Dynamic LDS layout for topk_kernel (float elements):
//   sbuf  : 2 * SLAB * ND     = 8192
//   skeys : 16 * ND           = 512
//   sqsq  : 16
//   l_val : NW*16*CAPK        = 6400
//   l_thr : NW*16             = 128
//   l_idx : NW*16*CAPK (int)  = 6400
//   l_mp  : NW*16 (int)       = 128
//   l_cnt : NW*16 (int)       = 128
#define TOPK_SMEM_ELEMS (8192 + 512 + 16 + 6400 + 128 + 6400 + 128 + 128)
#define TOPK_SMEM_BYTES (TOPK_SMEM_ELEMS * 4)

#ifndef HAVE_TDM
#if defined(__has_builtin)
#if __has_builtin(__builtin_amdgcn_tensor_load_to_lds) && \
    __has_builtin(__builtin_amdgcn_s_wait_tensorcnt)
#define HAVE_TDM 1
#endif
#endif
#endif
#ifndef HAVE_TDM
#define HAVE_TDM 0
#endif

__device__ __forceinline__ v8f v8f_zero() {
  v8f z = {0.f, 0.f, 0.f, 0.f, 0.f, 0.f, 0.f, 0.f};
  return z;
}

__device__ __forceinline__ v8f wmma_f32(v2f a, v2f b, v8f c) {
  // D = A(16x4,f32) * B(4x16,f32) + C(16x16,f32), full fp32
  return __builtin_amdgcn_wmma_f32_16x16x4_f32(
      false, a, false, b, (short)0, c, false, false);
}

__device__ __forceinline__ void wave_reduce_min(float &v, int &e) {
#pragma unroll
  for (int off = 16; off > 0; off >>= 1) {
    float ov = __shfl_xor(v, off, 32);
    int   oe = __shfl_xor(e, off, 32);
    if (ov < v || (ov == v && oe < e)) { v = ov; e = oe; }
  }
}

__device__ __forceinline__ void wave_reduce_max(float &v, int &e) {
#pragma unroll
  for (int off = 16; off > 0; off >>= 1) {
    float ov = __shfl_xor(v, off, 32);
    int   oe = __shfl_xor(e, off, 32);
    if (ov > v || (ov == v && oe < e)) { v = ov; e = oe; }
  }
}

#if HAVE_TDM
typedef unsigned int u32x4 __attribute__((ext_vector_type(4)));
typedef int i32x4 __attribute__((ext_vector_type(4)));
typedef int i32x8 __attribute__((ext_vector_type(8)));

// Issue a TDM 2-D tile load: `rows` rows of 32 f32 (contiguous, stride 32)
// from global `gaddr` into LDS byte-address `lds_addr`.  D# per ISA 8.3/8.4.
__device__ __forceinline__ void tdm_load_2d(unsigned lds_addr,
                                            const void* gaddr, int rows) {
  const unsigned long long ga = (unsigned long long)(uintptr_t)gaddr;
  u32x4 g0;
  g0[0] = 1u;                                    // count=1, user load
  g0[1] = lds_addr;                              // LDS dest (bytes)
  g0[2] = (unsigned)(ga & 0xFFFFFFFFu);          // global_addr[31:0]
  g0[3] = (unsigned)((ga >> 32) & 0x1FFFFFFu) |  // global_addr[56:32]
          (2u << 30);                            // type=2 ("image")
  i32x8 g1;
  g1[0] = (2 << 16);                             // data_size=2 -> 4 bytes
  g1[1] = (int)(32u << 16);                      // tensor_dim0 = 32
  g1[2] = (int)(((unsigned)NC & 0xFFFFu) << 16); // tensor_dim1 lo
  g1[3] = (int)(((unsigned)NC >> 16) |           // tensor_dim1 hi
                (32u << 16));                    // tile_dim0 = 32
  g1[4] = rows & 0xFFFF;                         // tile_dim1 = rows
  g1[5] = 32;                                    // tensor_dim0_stride = 32
  g1[6] = 0;
  g1[7] = 0;
  i32x4 z4 = {0, 0, 0, 0};
#if defined(__clang_major__) && (__clang_major__ >= 23)
  i32x8 z8 = {0, 0, 0, 0, 0, 0, 0, 0};
  __builtin_amdgcn_tensor_load_to_lds(g0, g1, z4, z4, z8, 0);
#else
  __builtin_amdgcn_tensor_load_to_lds(g0, g1, z4, z4, 0);
#endif
}
#endif  // HAVE_TDM

// ---------------------------------------------------------------------------
// Kernel 1: m_sq[a, c] = sum_d dnd_keys[a, c, d]^2
// ---------------------------------------------------------------------------
__global__ __launch_bounds__(256) void msq_kernel(const float* __restrict__ dnd,
                                                  float* __restrict__ msq) {
  int i = blockIdx.x * 256 + threadIdx.x;  // i in [0, NA*NC)
  const float4* p = (const float4*)(dnd + (size_t)i * ND);
  float s = 0.f;
#pragma unroll
  for (int j = 0; j < ND / 4; ++j) {
    float4 v = p[j];
    s += v.x * v.x + v.y * v.y + v.z * v.z + v.w * v.w;
  }
  msq[i] = s;
}

// ---------------------------------------------------------------------------
// Kernel 2: keys = relu(relu(x@W1 + b1)@W2 + b2) via chained f32 WMMAs.
// ---------------------------------------------------------------------------
__global__ __launch_bounds__(256) void mlp_kernel(
    const float* __restrict__ x, const float* __restrict__ W1,
    const float* __restrict__ b1, const float* __restrict__ W2,
    const float* __restrict__ b2, float* __restrict__ wkeys,
    float* __restrict__ okeys) {
  __shared__ float sh_h[NB * ND];

  const int tid  = threadIdx.x;
  const int w    = tid >> 5;
  const int lane = tid & 31;
  const int col  = lane & 15;
  const int half = lane >> 4;
  const int m0   = w * 16;

  v8f acc0 = v8f_zero(), acc1 = v8f_zero();
  for (int k0 = 0; k0 < NIN; k0 += 4) {
    const int kk = k0 + 2 * half;
    v2f a;  a.x  = x[(m0 + col) * NIN + kk];
            a.y  = x[(m0 + col) * NIN + kk + 1];
    v2f bA; bA.x = W1[kk * ND + col];
            bA.y = W1[(kk + 1) * ND + col];
    v2f bB; bB.x = W1[kk * ND + 16 + col];
            bB.y = W1[(kk + 1) * ND + 16 + col];
    acc0 = wmma_f32(a, bA, acc0);
    acc1 = wmma_f32(a, bB, acc1);
  }
#pragma unroll
  for (int r = 0; r < 8; ++r) {
    const int mm = r + 8 * half;
    sh_h[(m0 + mm) * ND + col]      = fmaxf(acc0[r] + b1[col], 0.f);
    sh_h[(m0 + mm) * ND + 16 + col] = fmaxf(acc1[r] + b1[16 + col], 0.f);
  }
  __syncthreads();

  acc0 = v8f_zero(); acc1 = v8f_zero();
#pragma unroll
  for (int k0 = 0; k0 < ND; k0 += 4) {
    const int kk = k0 + 2 * half;
    v2f a;  a.x  = sh_h[(m0 + col) * ND + kk];
            a.y  = sh_h[(m0 + col) * ND + kk + 1];
    v2f bA; bA.x = W2[kk * ND + col];
            bA.y = W2[(kk + 1) * ND + col];
    v2f bB; bB.x = W2[kk * ND + 16 + col];
            bB.y = W2[(kk + 1) * ND + 16 + col];
    acc0 = wmma_f32(a, bA, acc0);
    acc1 = wmma_f32(a, bB, acc1);
  }
#pragma unroll
  for (int r = 0; r < 8; ++r) {
    const int mm = r + 8 * half;
    const float k0v = fmaxf(acc0[r] + b2[col], 0.f);
    const float k1v = fmaxf(acc1[r] + b2[16 + col], 0.f);
    wkeys[(m0 + mm) * ND + col]      = k0v;
    wkeys[(m0 + mm) * ND + 16 + col] = k1v;
    okeys[(m0 + mm) * ND + col]      = k0v;
    okeys[(m0 + mm) * ND + 16 + col] = k1v;
  }
}

// ---------------------------------------------------------------------------
// Kernel 3: per (16-row b-tile, action) block.  TDM double-buffers 128x32
// dnd_keys slabs into LDS; waves run f32 WMMA cross tiles from LDS, maintain
// exact per-row top-50 lists, then merge and emit sorted indexes/scores + q.
// ---------------------------------------------------------------------------
__global__ __launch_bounds__(256) void topk_kernel(
    const float* __restrict__ dnd, const float* __restrict__ dvals,
    const float* __restrict__ msq, const float* __restrict__ keys,
    float* __restrict__ out, float* __restrict__ qout) {
  extern __shared__ float smem[];
  float* sbuf  = smem;                       // 2 * SLAB * ND
  float* skeys = sbuf + 2 * SLAB * ND;       // 16 * ND
  float* sqsq  = skeys + 16 * ND;            // 16
  float* l_val = sqsq + 16;                  // NW*16*CAPK
  float* l_thr = l_val + NW * 16 * CAPK;     // NW*16
  int*   l_idx = (int*)(l_thr + NW * 16);    // NW*16*CAPK
  int*   l_mp  = l_idx + NW * 16 * CAPK;     // NW*16
  int*   l_cnt = l_mp + NW * 16;             // NW*16

  const int aID  = blockIdx.x & 7;
  const int b0   = (blockIdx.x >> 3) * 16;
  const int tid  = threadIdx.x;
  const int w    = tid >> 5;
  const int lane = tid & 31;
  const int col  = lane & 15;
  const int half = lane >> 4;

  for (int i = tid; i < 16 * ND; i += 256) skeys[i] = keys[b0 * ND + i];
  if (tid < NW * 16) {
    l_thr[tid] = 3.0e38f;
    l_mp[tid]  = 0;
    l_cnt[tid] = 0;
  }
  __syncthreads();
  if (tid < 16) {
    float s = 0.f;
    for (int d = 0; d < ND; ++d) { float kv = skeys[tid * ND + d]; s += kv * kv; }
    sqsq[tid] = s;
  }
  __syncthreads();

  // Preload A operand (keys tile, constant for the whole scan).
  v2f areg[8];
#pragma unroll
  for (int kc = 0; kc < 8; ++kc) {
    const int kk = 4 * kc + 2 * half;
    areg[kc].x = skeys[col * ND + kk];
    areg[kc].y = skeys[col * ND + kk + 1];
  }
  float qs[8];
#pragma unroll
  for (int r = 0; r < 8; ++r) qs[r] = sqsq[r + 8 * half];

  const size_t abase = (size_t)aID * NC;
  const int nslabs = (NC + SLAB - 1) / SLAB;  // 782 (tail = 32 rows)

#if HAVE_TDM
  const unsigned lds0 = (unsigned)(uintptr_t)sbuf;
  const unsigned lds1 = (unsigned)(uintptr_t)(sbuf + SLAB * ND);
  if (w == 0) {
    tdm_load_2d(lds0, dnd + abase * ND, SLAB);
    tdm_load_2d(lds1, dnd + (abase + SLAB) * ND, SLAB);
    __builtin_amdgcn_s_wait_tensorcnt(1);  // slab 0 complete (in-order)
  }
  __syncthreads();
#endif

  for (int s = 0; s < nslabs; ++s) {
    const int cbase = s * SLAB;
    const int ccnt = (NC - cbase < SLAB) ? (NC - cbase) : SLAB;
    float* buf = sbuf + ((s & 1) * (SLAB * ND));

#if !HAVE_TDM
    for (int i = tid; i < ccnt * ND; i += 256)
      buf[i] = dnd[(abase + cbase) * ND + i];
    __syncthreads();
#endif

    if (w * 16 < ccnt) {  // wave-uniform guard (only tail slab trims waves)
      const int c0 = cbase + w * 16;
      const float* brow = buf + (w * 16 + col) * ND;

      v8f acc = v8f_zero();
#pragma unroll
      for (int kc = 0; kc < 8; ++kc) {
        const float2 bv = *(const float2*)(brow + 4 * kc + 2 * half);
        v2f b; b.x = bv.x; b.y = bv.y;
        acc = wmma_f32(areg[kc], b, acc);  // fp32 cross tile
      }
      const float msqv = msq[abase + c0 + col];

#pragma unroll
      for (int r = 0; r < 8; ++r) {
        const float v = qs[r] + msqv - 2.0f * acc[r];
        unsigned mask = (unsigned)__ballot(v < l_thr[w * 16 + r + 8 * half]);
        while (mask) {
          const int l = __ffs(mask) - 1;
          mask &= mask - 1;
          const float vl  = __shfl(v, l, 32);
          const int rowl  = r + ((l >> 4) << 3);
          const int lrow  = w * 16 + rowl;
          const int idxl  = c0 + (l & 15);
          const float thr = l_thr[lrow];
          if (vl < thr) {
            const int cnt = l_cnt[lrow];
            const int pos = (cnt < CAPK) ? cnt : l_mp[lrow];
            if (lane == 0) {
              l_val[lrow * CAPK + pos] = vl;
              l_idx[lrow * CAPK + pos] = idxl;
              if (cnt < CAPK) l_cnt[lrow] = cnt + 1;
            }
            const int newcnt = (cnt < CAPK) ? cnt + 1 : CAPK;
            if (newcnt == CAPK) {
              float mv = l_val[lrow * CAPK + lane];
              int   me = lane;
              if (lane + 32 < CAPK) {
                const float e2 = l_val[lrow * CAPK + lane + 32];
                if (e2 > mv) { mv = e2; me = lane + 32; }
              }
              wave_reduce_max(mv, me);
              if (lane == 0) { l_thr[lrow] = mv; l_mp[lrow] = me; }
            }
          }
        }
      }
    }
    __syncthreads();  // all waves done reading buf[s&1]

#if HAVE_TDM
    if (w == 0) {
      if (s + 2 < nslabs) {
        const int nb = cbase + 2 * SLAB;
        const int nrows = (NC - nb < SLAB) ? (NC - nb) : SLAB;
        tdm_load_2d((s & 1) ? lds1 : lds0, dnd + (abase + nb) * ND, nrows);
        __builtin_amdgcn_s_wait_tensorcnt(1);  // slab s+1 complete
      } else if (s + 1 < nslabs) {
        __builtin_amdgcn_s_wait_tensorcnt(0);  // drain last outstanding load
      }
    }
    __syncthreads();
#endif
  }
  __syncthreads();

  // Merge: wave w merges rows 2w, 2w+1 by iterated wave-min extraction
  // (sorted ascending d2 == jax.lax.top_k order on -d2).
  int* iout = (int*)out;
#pragma unroll 1
  for (int mi = 0; mi < 2; ++mi) {
    const int m = 2 * w + mi;
    const int b = b0 + m;
    const int obase = (b * NA + aID) * NP;
    float ssum = 0.f, qacc = 0.f;
#pragma unroll 1
    for (int p = 0; p < NP; ++p) {
      float mv = 3.0e38f;
      int   me = -1;
      for (int e = lane; e < NW * CAPK; e += 32) {
        const int ww = e / CAPK;
        const int pp = e - ww * CAPK;
        const float vv = l_val[(ww * 16 + m) * CAPK + pp];
        if (vv < mv) { mv = vv; me = e; }
      }
      wave_reduce_min(mv, me);
      const int ww = me / CAPK;
      const int pp = me - ww * CAPK;
      const int idx = l_idx[(ww * 16 + m) * CAPK + pp];
      if (lane == 0) l_val[(ww * 16 + m) * CAPK + pp] = 3.0e38f;  // consume
      const float score = 1.0f / (fmaxf(mv, 0.0f) + 0.001f);
      if (lane == 0) {
        out[OFF_SCORES + obase + p]   = score;
        iout[OFF_INDEXES + obase + p] = idx;
      }
      const float val = dvals[abase + idx];
      ssum += score;
      qacc += score * val;
    }
    if (lane == 0) qout[b * NA + aID] = qacc / ssum;
  }
}

// ---------------------------------------------------------------------------
// Kernel 4: values = max_a q[b,a]; actions = argmax_a (first occurrence).
// ---------------------------------------------------------------------------
__global__ __launch_bounds__(128) void finalize_kernel(const float* __restrict__ q,
                                                       float* __restrict__ out) {
  const int b = threadIdx.x;
  float best = q[b * NA];
  int bi = 0;
  for (int a = 1; a < NA; ++a) {
    const float qq = q[b * NA + a];
    if (qq > best) { best = qq; bi = a; }
  }
  out[OFF_VALUES + b] = best;
  ((int*)out)[OFF_ACTIONS + b] = bi;
}

// ---------------------------------------------------------------------------
extern "C" void kernel_launch(void* const* d_in, const int* in_sizes, int n_in,
                              void* d_out, int out_size, void* d_ws,
                              size_t ws_size, hipStream_t stream) {
  (void)in_sizes; (void)n_in; (void)out_size; (void)ws_size;
  const float* x     = (const float*)d_in[0];
  const float* W1    = (const float*)d_in[1];
  const float* b1    = (const float*)d_in[2];
  const float* W2    = (const float*)d_in[3];
  const float* b2    = (const float*)d_in[4];
  const float* dnd_k = (const float*)d_in[5];
  const float* dnd_v = (const float*)d_in[6];

  float* out   = (float*)d_out;
  float* ws    = (float*)d_ws;
  float* msq   = ws;                          // NA*NC = 800000 floats
  float* wkeys = ws + (size_t)NA * NC;        // 4096 floats
  float* wq    = wkeys + (size_t)NB * ND;     // 1024 floats

  msq_kernel<<<dim3((NA * NC) / 256), dim3(256), 0, stream>>>(dnd_k, msq);
  mlp_kernel<<<dim3(1), dim3(256), 0, stream>>>(x, W1, b1, W2, b2, wkeys,
                                                out + OFF_KEYS);
  topk_kernel<<<dim3((NB / 16) * NA), dim3(256), TOPK_SMEM_BYTES, stream>>>(
      dnd_k, dnd_v, msq, wkeys, out, wq);
  finalize_kernel<<<dim3(1), dim3(128), 0, stream>>>(wq, out);
}